// EncoderRNN_89000312308175
// MI455X (gfx1250) — compile-verified
//
#include <hip/hip_runtime.h>
#include <hip/hip_bf16.h>
#include <math.h>

typedef float v2f __attribute__((ext_vector_type(2)));
typedef float v4f __attribute__((ext_vector_type(4)));
typedef float v8f __attribute__((ext_vector_type(8)));

#define HID 1024
#define G3 (3 * HID)              // 3072 rows per matrix
#define NSPLIT 4                  // K-dimension split for extra wave parallelism
#define KSPLIT (HID / NSPLIT)     // 256 K per wave
#define ROWTILE 16
#define TILES_PER_MAT (G3 / ROWTILE)              // 192
#define WAVES_TOTAL (2 * TILES_PER_MAT * NSPLIT)  // 1536
#define WAVES_PER_BLOCK 8
#define NBLOCKS (WAVES_TOTAL / WAVES_PER_BLOCK)   // 192

// Phase 1: both GEMVs (w_ih@x, w_hh@h) via V_WMMA_F32_16X16X4_F32.
// Each wave: 16 rows x 256 K-slice; partial sums -> workspace.
__global__ __launch_bounds__(256) void gru_gemv_wmma(
    const int*   __restrict__ inp,      // [1] token index (int32 / low word of int64)
    const float* __restrict__ hidden,   // [HID]
    const float* __restrict__ emb,      // [VOCAB, HID]
    const float* __restrict__ w_ih,     // [3H, H]
    const float* __restrict__ w_hh,     // [3H, H]
    float*       __restrict__ partial)  // [NSPLIT, 2, 3H]
{
    __shared__ float vec[2 * HID];      // [0..H): x (embedded), [H..2H): h

    const int tid = threadIdx.x;

    // Stage x (OOV -> 0 via mask multiply, branch-free) and h into LDS.
    // 1024 floats / 256 threads = one b128 load + one b128 LDS store each.
    {
        const int v = inp[0];
        const float xscale = (v == -1) ? 0.0f : 1.0f;
        const long idx = (v < 0) ? 0 : (long)v;          // clamped: row 0 always valid
        v4f xv = *(const v4f*)(emb + idx * (long)HID + tid * 4);
        v4f hv = *(const v4f*)(hidden + tid * 4);
        xv *= xscale;
        *(v4f*)&vec[tid * 4]       = xv;
        *(v4f*)&vec[HID + tid * 4] = hv;
    }
    __syncthreads();

    const int wave  = blockIdx.x * WAVES_PER_BLOCK + (tid >> 5);
    const int lane  = tid & 31;
    const int split = wave & (NSPLIT - 1);
    const int rt    = wave >> 2;                     // row-tile id in [0, 384)
    const int m     = (rt >= TILES_PER_MAT) ? 1 : 0; // 0 = w_ih@x, 1 = w_hh@h
    const int rowBase = (rt - m * TILES_PER_MAT) * ROWTILE;
    const int kBase   = split * KSPLIT;
    const int halfSel = (lane >> 4) << 1;            // lanes 0-15: K{0,1}; 16-31: K{2,3}

    const float* __restrict__ W = m ? w_hh : w_ih;
    const float* aptr = W + (size_t)(rowBase + (lane & 15)) * HID + kBase + halfSel;
    const float* bptr = vec + m * HID + kBase + halfSel;

    v8f c = {};   // 16x16 f32 accumulator; all 16 columns identical (B broadcast)
    #pragma unroll 8
    for (int k = 0; k < KSPLIT; k += 4) {
        // A: 16x4 tile of W (b64 per lane; rows stream -> full cache-line use)
        v2f a = *(const v2f*)(aptr + k);
        // B: 4x16 broadcast of the vector segment (conflict-free LDS broadcast)
        v2f b = *(const v2f*)(bptr + k);
        c = __builtin_amdgcn_wmma_f32_16x16x4_f32(
                /*neg_a=*/false, a, /*neg_b=*/false, b,
                /*c_mod=*/(short)0, c, /*reuse_a=*/false, /*reuse_b=*/false);
    }

    // D layout: VGPR r = row (r | (lane>=16 ? 8 : 0)), every column equal.
    // Lanes 0 and 16 write the 16 partial row-sums (2x b128 each).
    if ((lane & 15) == 0) {
        const int rowOff = rowBase + ((lane >> 4) << 3);
        float* outp = partial + ((size_t)(split * 2 + m)) * G3 + rowOff;
        v4f lo = __builtin_shufflevector(c, c, 0, 1, 2, 3);
        v4f hi = __builtin_shufflevector(c, c, 4, 5, 6, 7);
        *(v4f*)(outp)     = lo;
        *(v4f*)(outp + 4) = hi;
    }
}

// Phase 2: reduce K-split partials, add biases, apply GRU gates, write h_new twice.
__global__ __launch_bounds__(256) void gru_combine(
    const float* __restrict__ hidden,
    const float* __restrict__ b_ih,
    const float* __restrict__ b_hh,
    const float* __restrict__ partial,  // [NSPLIT, 2, 3H]
    float*       __restrict__ out)      // [2 * HID] (tuple (out, out) flattened)
{
    const int i = blockIdx.x * blockDim.x + threadIdx.x;
    if (i >= HID) return;

    float s[6] = {0.f, 0.f, 0.f, 0.f, 0.f, 0.f}; // gi_r, gi_z, gi_n, gh_r, gh_z, gh_n
    #pragma unroll
    for (int sp = 0; sp < NSPLIT; ++sp) {
        const float* pi = partial + ((size_t)(sp * 2 + 0)) * G3;
        const float* ph = partial + ((size_t)(sp * 2 + 1)) * G3;
        s[0] += pi[i];            s[3] += ph[i];
        s[1] += pi[HID + i];      s[4] += ph[HID + i];
        s[2] += pi[2 * HID + i];  s[5] += ph[2 * HID + i];
    }
    const float gi_r = s[0] + b_ih[i];
    const float gi_z = s[1] + b_ih[HID + i];
    const float gi_n = s[2] + b_ih[2 * HID + i];
    const float gh_r = s[3] + b_hh[i];
    const float gh_z = s[4] + b_hh[HID + i];
    const float gh_n = s[5] + b_hh[2 * HID + i];

    const float r = 1.0f / (1.0f + expf(-(gi_r + gh_r)));
    const float z = 1.0f / (1.0f + expf(-(gi_z + gh_z)));
    const float n = tanhf(gi_n + r * gh_n);
    const float hn = (1.0f - z) * n + z * hidden[i];

    out[i]       = hn;
    out[HID + i] = hn;
}

extern "C" void kernel_launch(void* const* d_in, const int* in_sizes, int n_in,
                              void* d_out, int out_size, void* d_ws, size_t ws_size,
                              hipStream_t stream) {
    const int*   inp    = (const int*)  d_in[0];
    const float* hidden = (const float*)d_in[1];
    const float* emb    = (const float*)d_in[2];
    const float* w_ih   = (const float*)d_in[3];
    const float* w_hh   = (const float*)d_in[4];
    const float* b_ih   = (const float*)d_in[5];
    const float* b_hh   = (const float*)d_in[6];
    float* out     = (float*)d_out;
    float* partial = (float*)d_ws;     // needs NSPLIT*2*3072 floats = 96 KB

    gru_gemv_wmma<<<NBLOCKS, 256, 0, stream>>>(inp, hidden, emb, w_ih, w_hh, partial);
    gru_combine<<<(HID + 255) / 256, 256, 0, stream>>>(hidden, b_ih, b_hh, partial, out);
}